// UniBboxNet_52055003628273
// MI455X (gfx1250) — compile-verified
//
#include <hip/hip_runtime.h>

// Problem constants that cannot be recovered from flat input sizes
// (match the reference: feat_map is [B, C, 200, 304]).
#define FEAT_H 200
#define FEAT_W 304

// 8-byte corner-pair load with only 4-byte alignment guaranteed.
// __builtin_memcpy lets clang emit an align-4 8-byte global load without UB.
struct FPair { float lo, hi; };
__device__ __forceinline__ FPair load_pair4(const float* p) {
    FPair r;
    __builtin_memcpy(&r, p, sizeof(FPair));
    return r;
}

// One wave32 per sample point s = n*P + p.
// lane = channel within a 32-channel group; CHAN/32 groups, fully unrolled
// when CHAN is a compile-time constant -> 16 outstanding b64 gathers/wave.
template <int CHAN>
__global__ __launch_bounds__(256)
void deform_feats_kernel(const float* __restrict__ feat,
                         const float* __restrict__ rois,
                         const float* __restrict__ offs,
                         const int*   __restrict__ stride_p,
                         float*       __restrict__ out,
                         int NP, int P, int p_shift, int C_rt)
{
    const int C = (CHAN > 0) ? CHAN : C_rt;
    const int wavesPerBlock = blockDim.x >> 5;
    const int s = blockIdx.x * wavesPerBlock + (threadIdx.x >> 5);
    if (s >= NP) return;
    const int lane = threadIdx.x & 31;

    int n, p;
    if (p_shift >= 0) {            // P is a power of two (reference: P = 32)
        n = s >> p_shift;
        p = s & (P - 1);
    } else {
        n = s / P;
        p = s - n * P;
    }

    // ---- per-sample (wave-uniform) ROI / offset math --------------------
    const float* roi = rois + (size_t)n * 5;
    const int   b  = (int)roi[0];
    const float x1 = roi[1], y1 = roi[2], x2 = roi[3], y2 = roi[4];

    const float ox = offs[(size_t)n * (2 * P) + 2 * p + 0];
    const float oy = offs[(size_t)n * (2 * P) + 2 * p + 1];

    const float inv_stride = 1.0f / (float)(*stride_p);  // exact for pow2 stride

    const float cx = 0.5f * (x1 + x2);
    const float cy = 0.5f * (y1 + y2);
    const float bw = x2 - x1 + 1.0f;
    const float bh = y2 - y1 + 1.0f;

    const float w = (cx + ox * bw * 0.1f) * inv_stride;
    const float h = (cy + oy * bh * 0.1f) * inv_stride;

    // ---- bilinear setup, replicating reference border semantics ---------
    const float Hm1 = (float)(FEAT_H - 1);
    const float Wm1 = (float)(FEAT_W - 1);

    const float hl_f  = fminf(fmaxf(floorf(h), 0.0f), Hm1);
    const bool  hcap  = hl_f >= Hm1;
    const float hh_f  = hcap ? hl_f : hl_f + 1.0f;
    const float h_eff = hcap ? hl_f : h;
    const float lh  = h_eff - hl_f;
    const float hhw = 1.0f - lh;          // w3 = w4 = 0 when hcap

    const float wl_f  = fminf(fmaxf(floorf(w), 0.0f), Wm1);
    const bool  wcap  = wl_f >= Wm1;
    const float w_eff = wcap ? wl_f : w;
    const float lw = w_eff - wl_f;        // lw == 0 when wcap -> w2 = w4 = 0
    const float hw = 1.0f - lw;

    const float w1 = hhw * hw;            // (h_low,  w_low )
    const float w2 = hhw * lw;            // (h_low,  w_high)
    const float w3 = lh  * hw;            // (h_high, w_low )
    const float w4 = lh  * lw;            // (h_high, w_high)

    // Border handling via loop-invariant WEIGHT swap instead of per-load
    // data selects: pair is loaded at pw = wcap ? W-2 : wl, so under wcap the
    // wanted pixel (wl = W-1) is .hi and carries w1, while .lo carries w2==0.
    const float wLoA = wcap ? w2 : w1;    // weight of rowL pair .lo
    const float wHiA = wcap ? w1 : w2;    // weight of rowL pair .hi
    const float wLoB = wcap ? w4 : w3;    // weight of rowH pair .lo
    const float wHiB = wcap ? w3 : w4;    // weight of rowH pair .hi

    const int hl = (int)hl_f;
    const int hh = (int)hh_f;             // == hl when hcap (weights 0)
    const int wl = (int)wl_f;
    const int pw = wcap ? (FEAT_W - 2) : wl;   // never reads OOB

    const size_t HW = (size_t)FEAT_H * FEAT_W;
    const float* rowL = feat + ((size_t)b * C + lane) * HW + (size_t)hl * FEAT_W + pw;
    const float* rowH = feat + ((size_t)b * C + lane) * HW + (size_t)hh * FEAT_W + pw;
    float*       op   = out  + (size_t)s * C + lane;
    const size_t cstep = 32 * HW;         // channel-group stride in elements

    auto body = [&](int i) {
        const FPair a  = load_pair4(rowL + (size_t)i * cstep);
        const FPair bb = load_pair4(rowH + (size_t)i * cstep);
        float r = wLoA * a.lo;                 // pure 4-FMA chain, no selects
        r = fmaf(wHiA, a.hi, r);
        r = fmaf(wLoB, bb.lo, r);
        r = fmaf(wHiB, bb.hi, r);
        // Write-once streaming output: TH_NT store keeps the 192MB L2
        // dedicated to feat_map residency (feat_map fits entirely in L2).
        __builtin_nontemporal_store(r, op + (size_t)i * 32);
    };

    if (CHAN > 0) {
        constexpr int ITERS = (CHAN > 0) ? (CHAN / 32) : 1;
        #pragma unroll
        for (int i = 0; i < ITERS; ++i) body(i);   // full unroll: 16 loads in flight
    } else {
        const int iters = C >> 5;
        #pragma unroll 4
        for (int i = 0; i < iters; ++i) body(i);
    }
}

extern "C" void kernel_launch(void* const* d_in, const int* in_sizes, int n_in,
                              void* d_out, int out_size, void* d_ws, size_t ws_size,
                              hipStream_t stream)
{
    const float* feat     = (const float*)d_in[0];
    const float* rois     = (const float*)d_in[1];
    const float* offs     = (const float*)d_in[2];
    const int*   stride_p = (const int*)d_in[3];
    // d_in[4] = num_point (device scalar) — derived host-side instead.

    const int N  = in_sizes[1] / 5;            // rois: [N,5]
    const int P  = in_sizes[2] / (2 * N);      // offset: [N, 2P]
    const int NP = N * P;
    const int C  = out_size / NP;              // out: [N,P,C]

    // log2(P) when P is a power of two, else -1 (division fallback in-kernel)
    int p_shift = -1;
    if (P > 0 && (P & (P - 1)) == 0) {
        p_shift = 0;
        while ((1 << p_shift) < P) ++p_shift;
    }

    const int threads = 256;                   // 8 wave32 per block
    const int wpb     = threads / 32;
    const int blocks  = (NP + wpb - 1) / wpb;

    if (C == 256) {
        deform_feats_kernel<256><<<blocks, threads, 0, stream>>>(
            feat, rois, offs, stride_p, (float*)d_out, NP, P, p_shift, C);
    } else {
        deform_feats_kernel<0><<<blocks, threads, 0, stream>>>(
            feat, rois, offs, stride_p, (float*)d_out, NP, P, p_shift, C);
    }
}